// LinearAttention_82008105550007
// MI455X (gfx1250) — compile-verified
//
#include <hip/hip_runtime.h>

typedef float v2f __attribute__((ext_vector_type(2)));
typedef float v8f __attribute__((ext_vector_type(8)));

// V_WMMA_F32_16X16X4_F32: D(16x16 f32) = A(16x4) * B(4x16) + C   (codegen-confirmed round 1)
#define WMMA_F32(a_, b_, c_) \
    __builtin_amdgcn_wmma_f32_16x16x4_f32(false, (a_), false, (b_), (short)0, (c_), false, false)

// Low 32 bits of a flat pointer to LDS = LDS byte offset (aperture layout, ISA 10.2).
__device__ __forceinline__ unsigned ldsOff(const void* p) {
    return (unsigned)(uintptr_t)p;
}
// Async DMA: 16B global -> LDS, GVS addressing (SGPR64 base + VGPR32 byte offset).
__device__ __forceinline__ void asyncLd16(unsigned lds_byte, const float* sbase, int voff_byte) {
    asm volatile("global_load_async_to_lds_b128 %0, %1, %2"
                 :: "v"(lds_byte), "v"(voff_byte), "s"(sbase) : "memory");
}
__device__ __forceinline__ void waitAsync0() {
    asm volatile("s_wait_asynccnt 0x0" ::: "memory");
}

// Monotone float->uint key so row-max can use deterministic integer atomicMax.
__device__ __forceinline__ unsigned fkey(float f) {
    unsigned u = __float_as_uint(f);
    return (u & 0x80000000u) ? ~u : (u | 0x80000000u);
}
__device__ __forceinline__ float funkey(unsigned k) {
    return __uint_as_float((k & 0x80000000u) ? (k & 0x7FFFFFFFu) : ~k);
}

// ---------------------------------------------------------------- K0: init
__global__ void init_kernel(unsigned* kmaxKey) {
    kmaxKey[threadIdx.x] = 0u;   // 0 < key(any float)
}

// ---------------------------------------------------------------- K1: row max of k = W_k @ x
// Block: 32 positions. W_k (128x64) and the x tile staged via async DMA; each x byte
// loaded from global exactly once per block (was 8x redundant).
__global__ void kmax_kernel(const float* __restrict__ x, const float* __restrict__ Wqkv,
                            unsigned* __restrict__ kmaxKey, int N) {
    __shared__ float Wl[128][68];   // pad 68: 16B-aligned rows, conflict-free A-frag reads
    __shared__ float xt[64][36];    // pad 36: 16B-aligned rows
    const int bi = blockIdx.y;
    const int p0 = blockIdx.x * 32;
    const int tid = threadIdx.x;
    const int w = tid >> 5, L = tid & 31, lm = L & 15, lh = L >> 4;
    const float* xb = x + (size_t)bi * 64 * N;

    for (int i = tid; i < 2048; i += 256) {            // W_k slab: qkv rows 128..255
        const int r = i >> 4, sg = (i & 15) * 4;
        asyncLd16(ldsOff(&Wl[r][sg]), Wqkv, ((128 + r) * 64 + sg) * 4);
    }
    for (int i = tid; i < 512; i += 256) {             // x tile: 64 ch x 32 pos
        const int r = i >> 3, sg = (i & 7) * 4;
        asyncLd16(ldsOff(&xt[r][sg]), xb, (r * N + p0 + sg) * 4);
    }
    waitAsync0();
    __syncthreads();

    float mg[8];
    #pragma unroll
    for (int g = 0; g < 8; ++g) mg[g] = -3.4e38f;
    #pragma unroll
    for (int pt = 0; pt < 2; ++pt) {
        v8f acc = {};
        #pragma unroll
        for (int k0 = 0; k0 < 64; k0 += 4) {
            const int kk = k0 + lh * 2;
            v2f a, b;
            a.x = Wl[w * 16 + lm][kk];
            a.y = Wl[w * 16 + lm][kk + 1];
            b.x = xt[kk][pt * 16 + lm];
            b.y = xt[kk + 1][pt * 16 + lm];
            acc = WMMA_F32(a, b, acc);
        }
        #pragma unroll
        for (int g = 0; g < 8; ++g) mg[g] = fmaxf(mg[g], acc[g]);
    }
    #pragma unroll
    for (int g = 0; g < 8; ++g) {
        float m = mg[g];
        m = fmaxf(m, __shfl_xor(m, 1, 16));
        m = fmaxf(m, __shfl_xor(m, 2, 16));
        m = fmaxf(m, __shfl_xor(m, 4, 16));
        m = fmaxf(m, __shfl_xor(m, 8, 16));
        if (lm == 0)
            atomicMax(&kmaxKey[bi * 128 + w * 16 + g + lh * 8], fkey(m));
    }
}

// ---------------------------------------------------------------- K2: chunk partials of
// Z[d] = sum_n exp(k[d,n]-m_d)  and  C[d,e] = sum_n exp(k[d,n]-m_d)*v[e,n]
// One block per (1024-position chunk, head, batch); 32 sub-chunks of 32 positions.
__global__ void ctx_partial_kernel(const float* __restrict__ x, const float* __restrict__ Wqkv,
                                   const unsigned* __restrict__ kmaxKey,
                                   float* __restrict__ Zpart, float* __restrict__ Cpart,
                                   int N, int nchunk) {
    __shared__ float Wl[64][68];    // rows 0..31 = W_k head slab, 32..63 = W_v head slab
    __shared__ float xt[64][36];    // staged x sub-tile
    __shared__ float kv[64][36];    // rows 0..31 = k->exp(k), rows 32..63 = v
    __shared__ float kmaxf[32];
    const int chunk = blockIdx.x, h = blockIdx.y, bi = blockIdx.z;
    const int tid = threadIdx.x;
    const int w = tid >> 5, L = tid & 31, lm = L & 15, lh = L >> 4;
    const float* xb = x + (size_t)bi * 64 * N;

    for (int i = tid; i < 1024; i += 256) {            // stage k,v weight slab for head h
        const int r = i >> 4, sg = (i & 15) * 4;
        const int qrow = (r < 32) ? (128 + h * 32 + r) : (256 + h * 32 + (r - 32));
        asyncLd16(ldsOff(&Wl[r][sg]), Wqkv, (qrow * 64 + sg) * 4);
    }
    if (tid < 32) kmaxf[tid] = funkey(kmaxKey[bi * 128 + h * 32 + tid]);

    const int rt = w & 3, pt = w >> 2;                 // stage-1 tile for this wave
    v8f acc3 = {};                                     // context accumulator (waves 0..3)
    float zreg = 0.0f;                                 // Z accumulator (tid < 32)

    for (int sub = 0; sub < 32; ++sub) {
        const int pbase = chunk * 1024 + sub * 32;
        for (int i = tid; i < 512; i += 256) {         // stage x sub-tile
            const int r = i >> 3, sg = (i & 7) * 4;
            asyncLd16(ldsOff(&xt[r][sg]), xb, (r * N + pbase + sg) * 4);
        }
        waitAsync0();                                  // also covers Wl on first iter
        __syncthreads();
        // ---- stage 1: k,v = Wl @ xt via WMMA (8 waves x 1 tile of 16x16)
        v8f c1 = {};
        #pragma unroll
        for (int k0 = 0; k0 < 64; k0 += 4) {
            const int kk = k0 + lh * 2;
            v2f a, b;
            a.x = Wl[rt * 16 + lm][kk];
            a.y = Wl[rt * 16 + lm][kk + 1];
            b.x = xt[kk][pt * 16 + lm];
            b.y = xt[kk + 1][pt * 16 + lm];
            c1 = WMMA_F32(a, b, c1);
        }
        #pragma unroll
        for (int g = 0; g < 8; ++g)
            kv[rt * 16 + g + lh * 8][pt * 16 + lm] = c1[g];
        __syncthreads();
        // ---- stage 2: E = exp(k - rowmax), in place (rows 0..31)
        for (int i = tid; i < 1024; i += 256) {
            const int r = i >> 5, p = i & 31;
            kv[r][p] = __expf(kv[r][p] - kmaxf[r]);
        }
        __syncthreads();
        // ---- stage 3a: Z row sums (fixed serial order -> deterministic)
        if (tid < 32) {
            float s = 0.0f;
            #pragma unroll
            for (int p = 0; p < 32; ++p) s += kv[tid][p];
            zreg += s;
        }
        // ---- stage 3b: C += E(32x32) @ V^T(32x32) via WMMA (waves 0..3, 2x2 tiles)
        if (w < 4) {
            const int d0 = (w & 1) * 16, e0 = (w >> 1) * 16;
            #pragma unroll
            for (int k0 = 0; k0 < 32; k0 += 4) {
                const int kk = k0 + lh * 2;
                v2f a, b;
                a.x = kv[d0 + lm][kk];            // A[M=d][K=p]
                a.y = kv[d0 + lm][kk + 1];
                b.x = kv[32 + e0 + lm][kk];       // B[K=p][N=e] = v[e][p]
                b.y = kv[32 + e0 + lm][kk + 1];
                acc3 = WMMA_F32(a, b, acc3);
            }
        }
        __syncthreads();                               // readers done before xt/kv overwrite
    }
    if (w < 4) {
        const size_t cb = ((size_t)(bi * 4 + h) * nchunk + chunk) * 1024;
        const int d0 = (w & 1) * 16, e0 = (w >> 1) * 16;
        #pragma unroll
        for (int g = 0; g < 8; ++g)
            Cpart[cb + (d0 + g + lh * 8) * 32 + e0 + lm] = acc3[g];
    }
    if (tid < 32)
        Zpart[((size_t)(bi * 4 + h) * nchunk + chunk) * 32 + tid] = zreg;
}

// ---------------------------------------------------------------- K3: reduce chunks, normalize
__global__ void ctx_reduce_kernel(const float* __restrict__ Zpart, const float* __restrict__ Cpart,
                                  float* __restrict__ ctx, int nchunk) {
    const int bh = blockIdx.x;
    const int tid = threadIdx.x;
    __shared__ float zinv[32];
    if (tid < 32) {
        float s = 0.0f;
        for (int ch = 0; ch < nchunk; ++ch)
            s += Zpart[((size_t)bh * nchunk + ch) * 32 + tid];
        zinv[tid] = 1.0f / s;
    }
    __syncthreads();
    for (int i = tid; i < 1024; i += 256) {
        float s = 0.0f;
        for (int ch = 0; ch < nchunk; ++ch)
            s += Cpart[((size_t)bh * nchunk + ch) * 1024 + i];
        ctx[bh * 1024 + i] = s * zinv[i >> 5];
    }
}

// ---------------------------------------------------------------- K4: fold into G_b (64x64)
// M[h*32+e][c] = scale * sum_d ctx[h,d,e]*Wq[h*32+d][c];  G_b = W_out(64x128) @ M(128x64)
__global__ void gmat_kernel(const float* __restrict__ ctx, const float* __restrict__ Wqkv,
                            const float* __restrict__ Wout, float* __restrict__ G) {
    const int bi = blockIdx.x;
    const int tid = threadIdx.x;
    __shared__ float M[128][64];
    const float scale = 0.17677669529663687f;   // 32^-0.5
    for (int i = tid; i < 8192; i += 256) {
        const int he = i >> 6, c = i & 63;
        const int h = he >> 5, e = he & 31;
        float s = 0.0f;
        #pragma unroll 8
        for (int d = 0; d < 32; ++d)
            s += ctx[((bi * 4 + h) * 32 + d) * 32 + e] * Wqkv[(h * 32 + d) * 64 + c];
        M[he][c] = s * scale;
    }
    __syncthreads();
    for (int i = tid; i < 4096; i += 256) {
        const int o = i >> 6, c = i & 63;
        float s = 0.0f;
        #pragma unroll 8
        for (int he = 0; he < 128; ++he)
            s += Wout[o * 128 + he] * M[he][c];
        G[bi * 4096 + i] = s;
    }
}

// ---------------------------------------------------------------- K5: y = G_b @ x + b_out
// The single mandatory-bandwidth pass (113 MB). Each x byte DMA'd to LDS once, then WMMA.
__global__ void out_kernel(const float* __restrict__ x, const float* __restrict__ G,
                           const float* __restrict__ bout, float* __restrict__ y, int N) {
    __shared__ float Gl[64][68];
    __shared__ float xt[64][36];
    const int bi = blockIdx.y;
    const int p0 = blockIdx.x * 32;
    const int tid = threadIdx.x;
    const int w = tid >> 5, L = tid & 31, lm = L & 15, lh = L >> 4;
    const int rt = w & 3, pt = w >> 2;
    const float* xb = x + (size_t)bi * 64 * N;
    const float* Gb = G + bi * 4096;

    for (int i = tid; i < 1024; i += 256) {            // stage G_b (64x64)
        const int r = i >> 4, sg = (i & 15) * 4;
        asyncLd16(ldsOff(&Gl[r][sg]), Gb, (r * 64 + sg) * 4);
    }
    for (int i = tid; i < 512; i += 256) {             // stage x tile (64 ch x 32 pos)
        const int r = i >> 3, sg = (i & 7) * 4;
        asyncLd16(ldsOff(&xt[r][sg]), xb, (r * N + p0 + sg) * 4);
    }
    waitAsync0();
    __syncthreads();

    v8f acc = {};
    #pragma unroll
    for (int k0 = 0; k0 < 64; k0 += 4) {
        const int kk = k0 + lh * 2;
        v2f a, b;
        a.x = Gl[rt * 16 + lm][kk];
        a.y = Gl[rt * 16 + lm][kk + 1];
        b.x = xt[kk][pt * 16 + lm];
        b.y = xt[kk + 1][pt * 16 + lm];
        acc = WMMA_F32(a, b, acc);
    }
    const int p = p0 + pt * 16 + lm;
    float* yb = y + (size_t)bi * 64 * N;
    #pragma unroll
    for (int g = 0; g < 8; ++g) {
        const int row = rt * 16 + g + lh * 8;
        yb[row * N + p] = acc[g] + bout[row];
    }
}

// ---------------------------------------------------------------- launch
extern "C" void kernel_launch(void* const* d_in, const int* in_sizes, int n_in,
                              void* d_out, int out_size, void* d_ws, size_t ws_size,
                              hipStream_t stream) {
    const float* x    = (const float*)d_in[0];   // [2,64,48,48,48]
    const float* Wqkv = (const float*)d_in[1];   // [384,64]
    const float* Wout = (const float*)d_in[2];   // [64,128]
    const float* bout = (const float*)d_in[3];   // [64]
    float* y = (float*)d_out;

    const int B = 2, C = 64;
    const int N = in_sizes[0] / (B * C);         // 110592
    const int nchunk = N / 1024;                 // 108

    // workspace layout (floats): ~3.7 MB total
    float* ws = (float*)d_ws;
    unsigned* kmaxKey = (unsigned*)ws;                                   // 256
    float* Zpart = ws + 256;                                             // B*4*nchunk*32
    float* Cpart = Zpart + (size_t)B * 4 * nchunk * 32;                  // B*4*nchunk*1024
    float* ctx   = Cpart + (size_t)B * 4 * nchunk * 1024;                // B*4096
    float* G     = ctx + (size_t)B * 4096;                               // B*4096

    init_kernel<<<1, 256, 0, stream>>>(kmaxKey);
    kmax_kernel<<<dim3(N / 32, B), 256, 0, stream>>>(x, Wqkv, kmaxKey, N);
    ctx_partial_kernel<<<dim3(nchunk, 4, B), 256, 0, stream>>>(x, Wqkv, kmaxKey, Zpart, Cpart, N, nchunk);
    ctx_reduce_kernel<<<dim3(B * 4), 256, 0, stream>>>(Zpart, Cpart, ctx, nchunk);
    gmat_kernel<<<dim3(B), 256, 0, stream>>>(ctx, Wqkv, Wout, G);
    out_kernel<<<dim3(N / 32, B), 256, 0, stream>>>(x, G, bout, y, N);
}